// BC_sPool_77137612636468
// MI455X (gfx1250) — compile-verified
//
#include <hip/hip_runtime.h>

typedef float v2f  __attribute__((ext_vector_type(2)));
typedef float v8f  __attribute__((ext_vector_type(8)));
typedef unsigned int u32x4 __attribute__((ext_vector_type(4)));
typedef int   i32x4 __attribute__((ext_vector_type(4)));
typedef int   i32x8 __attribute__((ext_vector_type(8)));

#define PI_F 3.14159265358979323846f

// D(16x16) += A(16x4) * B(4x16), fp32 WMMA (CDNA5)
__device__ __forceinline__ v8f wmma4(v2f a, v2f b, v8f c) {
  return __builtin_amdgcn_wmma_f32_16x16x4_f32(false, a, false, b, (short)0, c,
                                               false, false);
}

// ---------------------------------------------------------------------------
// TDM: async-load two 16KB planes (real at gsrc, imag at gsrc + 128*4096 floats)
// into LDS as one 2-D tensor op: tile 4096 x 2 elements of 4B, row stride
// 128*4096 elements. Issued by one wave; completion via TENSORcnt.
// Descriptor layout per CDNA5 ISA 8.3/8.4.
// ---------------------------------------------------------------------------
__device__ __forceinline__ void tdm_load_2planes(const float* gsrc,
                                                 unsigned ldsByteOff) {
  unsigned long long ga = (unsigned long long)(uintptr_t)gsrc;
  u32x4 g0;
  g0[0] = 1u;                                         // count=1, user, no gather
  g0[1] = ldsByteOff;                                 // lds_addr (bytes)
  g0[2] = (unsigned)(ga & 0xFFFFFFFFull);             // global_addr[31:0]
  g0[3] = (unsigned)((ga >> 32) & 0x01FFFFFFull)      // global_addr[56:32]
          | 0x80000000u;                              // type = 2 ("image")
  i32x8 g1;
  g1[0] = (2 << 16);                                  // data_size = 4B
  g1[1] = (int)(4096u << 16);                         // tensor_dim0 lo16
  g1[2] = (int)((4096u >> 16) | (2u << 16));          // dim0 hi16 | tensor_dim1 lo16
  g1[3] = (int)(4096u << 16);                         // dim1 hi16(0) | tile_dim0=4096
  g1[4] = 2;                                          // tile_dim1=2 | tile_dim2=0
  g1[5] = (int)(128u * 4096u);                        // tensor_dim0_stride[31:0]
  g1[6] = 0;                                          // stride0 hi | stride1 lo
  g1[7] = 0;                                          // stride1 hi
  i32x4 gz4 = {0, 0, 0, 0};                           // groups 2/3 unused (2-D)
#if defined(__clang_major__) && __clang_major__ >= 23
  i32x8 gz8 = {0, 0, 0, 0, 0, 0, 0, 0};
  __builtin_amdgcn_tensor_load_to_lds(g0, g1, gz4, gz4, gz8, 0);
#else
  __builtin_amdgcn_tensor_load_to_lds(g0, g1, gz4, gz4, 0);
#endif
}

// ---------------------------------------------------------------------------
// Build the fused spectral-pool operator A (32x64 complex):
//   A = IFFT32_ortho * CropRows * FFT64_ortho
//   A[m,h] = 1/(8*sqrt(32)) * sum_{p=0..31} e^{+2pi i m p/32} e^{-2pi i k(p) h/64}
//   k(p) = p (p<16) else p+32.  Exponent period 64 -> reduce mod 64.
// ---------------------------------------------------------------------------
__global__ void build_A_kernel(float* __restrict__ wsA) {
  int idx = blockIdx.x * blockDim.x + threadIdx.x;
  if (idx >= 2048) return;
  int m = idx >> 6;
  int h = idx & 63;
  float re = 0.f, im = 0.f;
  for (int p = 0; p < 32; ++p) {
    int k = (p < 16) ? p : (p + 32);
    int t = (2 * m * p - k * h) & 63;
    float ang = PI_F * (1.0f / 32.0f) * (float)t;
    re += __cosf(ang);
    im += __sinf(ang);
  }
  const float s = 1.0f / (8.0f * sqrtf(32.0f));
  wsA[idx]        = s * re;
  wsA[2048 + idx] = s * im;
}

// ---------------------------------------------------------------------------
// Per image (b,c): Y(32x32, complex) = A(32x64) * X(64x64) * A^T via
// V_WMMA_F32_16X16X4_F32. One block per image, 8 waves, X staged by TDM.
// ---------------------------------------------------------------------------
__global__ __launch_bounds__(256)
void spool_kernel(const float* __restrict__ x, const float* __restrict__ wsA,
                  float* __restrict__ out) {
  const int img = blockIdx.x;            // 0..4095
  const int b   = img >> 7;
  const int c   = img & 127;
  const float* xr = x + ((size_t)(b * 256 + c)) * 4096;  // imag plane at +128*4096

  __shared__ float sX[2 * 4096];         // [0]=real, [4096]=imag, dense stride 64
  __shared__ float sAr[32 * 65];         // A real, padded
  __shared__ float sAi[32 * 65];         // A imag
  __shared__ float sAn[32 * 65];         // -A imag (f32 WMMA has no A-neg)

  const int tid  = threadIdx.x;
  const int lane = tid & 31;
  const int wave = tid >> 5;             // 0..7
  const int half = lane >> 4;
  const int ln   = lane & 15;

  // Async tensor DMA of both X planes into LDS (wave 0 issues + waits)
  if (wave == 0) {
    tdm_load_2planes(xr, (unsigned)(uintptr_t)(void*)sX);
  }
  // Stage A into LDS with +1 padding, plus negated-imag copy
  for (int i = tid; i < 2048; i += 256) {
    int m = i >> 6, h = i & 63;
    float ar = wsA[i], ai = wsA[2048 + i];
    sAr[m * 65 + h] =  ar;
    sAi[m * 65 + h] =  ai;
    sAn[m * 65 + h] = -ai;
  }
  if (wave == 0) {
    __builtin_amdgcn_s_wait_tensorcnt(0);   // TDM writes to LDS complete
  }
  __syncthreads();

  // ---- Stage 1: T(32x64) = A * X (complex). 8 tiles of 16x16, 1 per wave ----
  const int m0 = (wave >> 2) << 4;
  const int n0 = (wave & 3)  << 4;
  v8f accR = {0.f,0.f,0.f,0.f,0.f,0.f,0.f,0.f};
  v8f accI = {0.f,0.f,0.f,0.f,0.f,0.f,0.f,0.f};
#pragma unroll
  for (int k = 0; k < 64; k += 4) {
    const int ka   = k + 2 * half;       // lanes 0-15: k,k+1 ; 16-31: k+2,k+3
    const int aoff = (m0 + ln) * 65 + ka;
    v2f aR = { sAr[aoff], sAr[aoff + 1] };
    v2f aI = { sAi[aoff], sAi[aoff + 1] };
    v2f aN = { sAn[aoff], sAn[aoff + 1] };
    v2f bR = { sX[ka * 64 + n0 + ln],        sX[(ka + 1) * 64 + n0 + ln] };
    v2f bI = { sX[4096 + ka * 64 + n0 + ln], sX[4096 + (ka + 1) * 64 + n0 + ln] };
    accR = wmma4(aR, bR, accR);          // + Ar*Xr
    accR = wmma4(aN, bI, accR);          // - Ai*Xi
    accI = wmma4(aR, bI, accI);          // + Ar*Xi
    accI = wmma4(aI, bR, accI);          // + Ai*Xr
  }
  __syncthreads();                       // all waves done reading X

  // Park T in X's LDS space, padded stride 65 (fixes A-operand column reads)
  float* sTr = sX;
  float* sTi = sX + 4096;
#pragma unroll
  for (int r = 0; r < 8; ++r) {
    int row = m0 + r + 8 * half;         // D layout: VGPR r -> row r + 8*half
    sTr[row * 65 + n0 + ln] = accR[r];
    sTi[row * 65 + n0 + ln] = accI[r];
  }
  __syncthreads();

  // ---- Stage 2: Y(32x32) = T * A^T (complex). 4 tiles x {re,im} ----
  const int tile = wave & 3;
  const int comp = wave >> 2;            // 0 = real, 1 = imag
  const int ym0  = (tile >> 1) << 4;
  const int yw0  = (tile & 1)  << 4;
  const float* bA0 = (comp == 0) ? sAr : sAi;  // pairs with Tr
  const float* bA1 = (comp == 0) ? sAn : sAr;  // pairs with Ti
  v8f acc = {0.f,0.f,0.f,0.f,0.f,0.f,0.f,0.f};
#pragma unroll
  for (int k = 0; k < 64; k += 4) {
    const int ka   = k + 2 * half;
    const int toff = (ym0 + ln) * 65 + ka;
    v2f tR = { sTr[toff], sTr[toff + 1] };
    v2f tI = { sTi[toff], sTi[toff + 1] };
    const int boff = (yw0 + ln) * 65 + ka;     // B[kk][j] = A[yw0+j][kk]
    v2f b0 = { bA0[boff], bA0[boff + 1] };
    v2f b1 = { bA1[boff], bA1[boff + 1] };
    acc = wmma4(tR, b0, acc);
    acc = wmma4(tI, b1, acc);
  }

  // Write Y tile: real -> channel c, imag -> channel c+128
  const int cOut = c + comp * 128;
  float* yp = out + (((size_t)(b * 256 + cOut)) * 32 + ym0) * 32;
#pragma unroll
  for (int r = 0; r < 8; ++r) {
    int row = r + 8 * half;
    yp[(size_t)row * 32 + yw0 + ln] = acc[r];
  }
}

extern "C" void kernel_launch(void* const* d_in, const int* in_sizes, int n_in,
                              void* d_out, int out_size, void* d_ws, size_t ws_size,
                              hipStream_t stream) {
  (void)in_sizes; (void)n_in; (void)out_size; (void)ws_size;
  const float* x = (const float*)d_in[0];
  float* wsA = (float*)d_ws;             // 4096 floats (16 KB) used
  float* out = (float*)d_out;

  build_A_kernel<<<8, 256, 0, stream>>>(wsA);
  spool_kernel<<<4096, 256, 0, stream>>>(x, wsA, out);
}